// S4Model_SS5_67619965108692
// MI455X (gfx1250) — compile-verified
//
#include <hip/hip_runtime.h>
#include <math.h>

typedef __attribute__((ext_vector_type(2))) float v2f;
typedef __attribute__((ext_vector_type(8))) float v8f;

#define SRATE 20000.0f
#define KS 251
#define HDIM 256
#define NMODES 32
#define NLAYERS 4
#define BATCH 32
#define LSEQ 735
#define PI_F 3.14159265358979323846f
#define BN_INV 0.9999950000374997f  /* 1/sqrt(1+1e-5) */

#define WMMA_F32(A, Bv, C) \
  __builtin_amdgcn_wmma_f32_16x16x4_f32(false, (A), false, (Bv), (short)0, (C), false, false)

// ---------------------------------------------------------------------------
// Sinc filter construction (16 x 251)
// ---------------------------------------------------------------------------
__global__ void k_build_sinc_filters(const float* __restrict__ low_hz,
                                     const float* __restrict__ band_hz,
                                     float* __restrict__ filt) {
  int idx = blockIdx.x * blockDim.x + threadIdx.x;
  if (idx >= 16 * KS) return;
  int i = idx / KS, j = idx % KS;
  float low = 50.0f + fabsf(low_hz[i]);
  float high = low + 50.0f + fabsf(band_hz[i]);
  high = fminf(fmaxf(high, 50.0f), SRATE * 0.5f);
  float band = high - low;
  float val;
  if (j == 125) {
    val = 1.0f;  // center = 2*band / (2*band)
  } else {
    int jj = (j < 125) ? j : 250 - j;       // symmetric
    float n_ = 2.0f * PI_F * (float)(jj - 125) / SRATE;
    float nlin = (float)jj * (124.5f / 124.0f);
    float window = 0.54f - 0.46f * cosf(2.0f * PI_F * nlin / (float)KS);
    float left = (sinf(high * n_) - sinf(low * n_)) / (n_ * 0.5f) * window;
    val = left / (2.0f * band);
  }
  filt[i * KS + j] = val;
}

// ---------------------------------------------------------------------------
// Sinc conv: (B,1,30250) -> (B,16,30000), LDS-staged input tile + filter row
// ---------------------------------------------------------------------------
__global__ void k_sinc_conv(const float* __restrict__ x,
                            const float* __restrict__ filt,
                            float* __restrict__ out) {
  __shared__ float sf[KS];
  __shared__ float sx[256 + KS - 1];
  int bc = blockIdx.y;               // b*16 + c
  int b = bc >> 4, c = bc & 15;
  int t0 = blockIdx.x * 256;
  int tid = threadIdx.x;
  for (int q = tid; q < KS; q += 256) sf[q] = filt[c * KS + q];
  const float* xb = x + b * 30250;
  for (int q = tid; q < 256 + KS - 1; q += 256) {
    int xi = t0 + q;
    sx[q] = (xi < 30250) ? xb[xi] : 0.0f;
  }
  __syncthreads();
  int t = t0 + tid;
  if (t >= 30000) return;
  float acc = 0.0f;
#pragma unroll 8
  for (int q = 0; q < KS; ++q) acc += sf[q] * sx[tid + q];
  out[(long)bc * 30000 + t] = acc;
}

// ---------------------------------------------------------------------------
// Conv1d + BN(eval) + PReLU as implicit GEMM on V_WMMA_F32_16X16X4_F32.
// One wave computes a 32(Cout) x 64(positions) tile: 2 A fragments x 4 B
// fragments -> 12 loads per 8 WMMAs. K->(cin,tap) mapping is strength-
// reduced to incremental 32-bit offsets; interior N-tiles use immediate-
// offset B loads (4 loads off one per-lane address).
// ---------------------------------------------------------------------------
template <int KW, int STRIDE>
__global__ void k_conv_bn_prelu_wmma(const float* __restrict__ x,
                                     const float* __restrict__ w,
                                     const float* __restrict__ bnw,
                                     const float* __restrict__ bnb,
                                     const float* __restrict__ prelu, int pidx,
                                     float* __restrict__ out,
                                     int Cin, int Lin, int Cout, int Lout) {
  int warp = threadIdx.x >> 5;
  int lane = threadIdx.x & 31;
  int tilesN = (Lout + 63) >> 6;
  int tilesM = Cout >> 5;            // 32-row M tiles
  int wid = blockIdx.x * 8 + warp;
  int total = BATCH * tilesM * tilesN;
  if (wid >= total) return;          // wave-uniform: EXEC stays all-1s
  int tn = wid % tilesN;
  int tm = (wid / tilesN) % tilesM;
  int b = wid / (tilesN * tilesM);
  int half = lane >> 4, l16 = lane & 15;
  int m0 = tm * 32, n0 = tn * 64;
  int KT = Cin * KW;
  const float* xb = x + (long)b * Cin * Lin;
  const float* wm = w + (long)(m0 + l16) * KT;
  const float* wm2 = wm + (long)16 * KT;
  __builtin_prefetch(wm, 0, 0);
  __builtin_prefetch(wm2, 0, 0);

  // incremental K -> (cin, tap) state for the two K-slots of this half-lane
  int ka0 = half * 2, ka1 = ka0 + 1;              // < 4 < KW
  int q0 = ka0, q1 = ka1;
  int off0 = ka0, off1 = ka1;                      // cin=0 initially

  v8f a00 = {}, a01 = {}, a02 = {}, a03 = {};
  v8f a10 = {}, a11 = {}, a12 = {}, a13 = {};

  if (n0 + 63 < Lout) {  // -------- interior tile: imm-offset B loads --------
    int nj0 = (n0 + l16) * STRIDE;
    int woff = 0;
    for (int k0 = 0; k0 < KT; k0 += 4) {
      v2f A0, A1;
      A0.x = wm[woff];  A0.y = wm[woff + 1];
      A1.x = wm2[woff]; A1.y = wm2[woff + 1];
      const float* p0 = xb + (off0 + nj0);
      const float* p1 = xb + (off1 + nj0);
      v2f b0, b1, b2, b3;
      b0.x = p0[0];           b0.y = p1[0];
      b1.x = p0[16 * STRIDE]; b1.y = p1[16 * STRIDE];
      b2.x = p0[32 * STRIDE]; b2.y = p1[32 * STRIDE];
      b3.x = p0[48 * STRIDE]; b3.y = p1[48 * STRIDE];
      a00 = WMMA_F32(A0, b0, a00); a01 = WMMA_F32(A0, b1, a01);
      a02 = WMMA_F32(A0, b2, a02); a03 = WMMA_F32(A0, b3, a03);
      a10 = WMMA_F32(A1, b0, a10); a11 = WMMA_F32(A1, b1, a11);
      a12 = WMMA_F32(A1, b2, a12); a13 = WMMA_F32(A1, b3, a13);
      woff += 4;
      q0 += 4; if (q0 >= KW) { q0 -= KW; off0 += Lin - KW; } off0 += 4;
      q1 += 4; if (q1 >= KW) { q1 -= KW; off1 += Lin - KW; } off1 += 4;
    }
  } else {               // -------- edge tile: clamped per-subtile offsets ---
    int nj[4];
#pragma unroll
    for (int j = 0; j < 4; ++j) {
      int n = n0 + 16 * j + l16;
      if (n > Lout - 1) n = Lout - 1;
      nj[j] = n * STRIDE;
    }
    int woff = 0;
    for (int k0 = 0; k0 < KT; k0 += 4) {
      v2f A0, A1;
      A0.x = wm[woff];  A0.y = wm[woff + 1];
      A1.x = wm2[woff]; A1.y = wm2[woff + 1];
      v2f b0, b1, b2, b3;
      b0.x = xb[off0 + nj[0]]; b0.y = xb[off1 + nj[0]];
      b1.x = xb[off0 + nj[1]]; b1.y = xb[off1 + nj[1]];
      b2.x = xb[off0 + nj[2]]; b2.y = xb[off1 + nj[2]];
      b3.x = xb[off0 + nj[3]]; b3.y = xb[off1 + nj[3]];
      a00 = WMMA_F32(A0, b0, a00); a01 = WMMA_F32(A0, b1, a01);
      a02 = WMMA_F32(A0, b2, a02); a03 = WMMA_F32(A0, b3, a03);
      a10 = WMMA_F32(A1, b0, a10); a11 = WMMA_F32(A1, b1, a11);
      a12 = WMMA_F32(A1, b2, a12); a13 = WMMA_F32(A1, b3, a13);
      woff += 4;
      q0 += 4; if (q0 >= KW) { q0 -= KW; off0 += Lin - KW; } off0 += 4;
      q1 += 4; if (q1 >= KW) { q1 -= KW; off1 += Lin - KW; } off1 += 4;
    }
  }

  float a = prelu[pidx];
  v8f* accs[8] = {&a00, &a01, &a02, &a03, &a10, &a11, &a12, &a13};
#pragma unroll
  for (int i = 0; i < 2; ++i) {
#pragma unroll
    for (int j = 0; j < 4; ++j) {
      int nn = n0 + 16 * j + l16;
      if (nn < Lout) {
#pragma unroll
        for (int r = 0; r < 8; ++r) {
          int mm = m0 + i * 16 + r + half * 8;
          float v = (*accs[i * 4 + j])[r] * (bnw[mm] * BN_INV) + bnb[mm];
          out[((long)b * Cout + mm) * Lout + nn] = (v >= 0.0f) ? v : a * v;
        }
      }
    }
  }
}

// ---------------------------------------------------------------------------
// S4D kernel build: K[h,l] = 2*Re( sum_n Ct[h,n] * exp(dtA[h,n]*l) )
// ---------------------------------------------------------------------------
__global__ void k_s4d_build(const float* __restrict__ log_dt,
                            const float* __restrict__ Cri,
                            const float* __restrict__ logA,
                            const float* __restrict__ Aimag,
                            float* __restrict__ Kf, int L) {
  __shared__ float ctr[NMODES], cti[NMODES], dre[NMODES], dimg[NMODES];
  int hh = blockIdx.x;
  int tid = threadIdx.x;
  if (tid < NMODES) {
    float dt = __expf(log_dt[hh]);
    float are = -__expf(logA[hh * NMODES + tid]);
    float aim = Aimag[hh * NMODES + tid];
    float dr = are * dt, di = aim * dt;
    float er = __expf(dr);
    float sn, cs;
    __sincosf(di, &sn, &cs);
    float numr = er * cs - 1.0f, numi = er * sn;  // exp(dtA) - 1
    float den = are * are + aim * aim;            // |A|^2
    float tr = (numr * are + numi * aim) / den;   // num / A
    float ti = (numi * are - numr * aim) / den;
    float cre = Cri[(hh * NMODES + tid) * 2 + 0];
    float cim = Cri[(hh * NMODES + tid) * 2 + 1];
    ctr[tid] = cre * tr - cim * ti;
    cti[tid] = cre * ti + cim * tr;
    dre[tid] = dr;
    dimg[tid] = di;
  }
  __syncthreads();
  for (int l = tid; l < L; l += blockDim.x) {
    float fl = (float)l;
    float acc = 0.0f;
#pragma unroll
    for (int nn = 0; nn < NMODES; ++nn) {
      float r = __expf(dre[nn] * fl);
      float sn, cs;
      __sincosf(dimg[nn] * fl, &sn, &cs);
      acc += r * (ctr[nn] * cs - cti[nn] * sn);
    }
    Kf[(long)hh * L + l] = 2.0f * acc;
  }
}

// ---------------------------------------------------------------------------
// irfft(rfft(u,2L)*rfft(k,2L))[:L] == causal conv. + D skip + exact GELU.
// Block per (b,h); u row and k row staged in LDS.
// ---------------------------------------------------------------------------
__global__ void k_causal_conv_gelu(const float* __restrict__ h,
                                   const float* __restrict__ Kf,
                                   const float* __restrict__ D,
                                   float* __restrict__ z, int L) {
  __shared__ float su[LSEQ], sk[LSEQ];
  int bh = blockIdx.x;
  int hc = bh % HDIM;
  const float* ur = h + (long)bh * L;
  const float* kr = Kf + (long)hc * L;
  for (int t = threadIdx.x; t < L; t += blockDim.x) {
    su[t] = ur[t];
    sk[t] = kr[t];
  }
  __syncthreads();
  float Dh = D[hc];
  for (int t = threadIdx.x; t < L; t += blockDim.x) {
    float acc = Dh * su[t];
    for (int s = 0; s <= t; ++s) acc += sk[s] * su[t - s];
    z[(long)bh * L + t] = 0.5f * acc * (1.0f + erff(acc * 0.7071067811865475f));
  }
}

// ---------------------------------------------------------------------------
// Y[b] = W (M x K) @ Z[b] (K x N) + bias, f32 WMMA, 32x64 tile per wave.
// M=512, K=256, N=735.
// ---------------------------------------------------------------------------
__global__ void k_gemm_wmma_bias(const float* __restrict__ W,
                                 const float* __restrict__ Z,
                                 const float* __restrict__ bias,
                                 float* __restrict__ Y,
                                 int M, int K, int N, int Bn) {
  int warp = threadIdx.x >> 5;
  int lane = threadIdx.x & 31;
  int tilesN = (N + 63) >> 6;
  int tilesM = M >> 5;
  int wid = blockIdx.x * 8 + warp;
  int total = Bn * tilesM * tilesN;
  if (wid >= total) return;          // wave-uniform
  int tn = wid % tilesN;
  int tm = (wid / tilesN) % tilesM;
  int b = wid / (tilesN * tilesM);
  int half = lane >> 4, l16 = lane & 15;
  int m0 = tm * 32, n0 = tn * 64;
  const float* Zb = Z + (long)b * K * N;
  const float* Wm = W + (long)(m0 + l16) * K;
  const float* Wm2 = Wm + (long)16 * K;
  __builtin_prefetch(Wm, 0, 0);
  __builtin_prefetch(Wm2, 0, 0);

  v8f a00 = {}, a01 = {}, a02 = {}, a03 = {};
  v8f a10 = {}, a11 = {}, a12 = {}, a13 = {};

  if (n0 + 63 < N) {  // -------- interior: imm-offset B loads --------
    int bidx = (half * 2) * N + n0 + l16;  // row ka, col n
    int woff = 0;
    for (int k0 = 0; k0 < K; k0 += 4) {
      v2f A0, A1;
      A0.x = Wm[woff];  A0.y = Wm[woff + 1];
      A1.x = Wm2[woff]; A1.y = Wm2[woff + 1];
      const float* p0 = Zb + bidx;
      const float* p1 = p0 + N;
      v2f b0, b1, b2, b3;
      b0.x = p0[0];  b0.y = p1[0];
      b1.x = p0[16]; b1.y = p1[16];
      b2.x = p0[32]; b2.y = p1[32];
      b3.x = p0[48]; b3.y = p1[48];
      a00 = WMMA_F32(A0, b0, a00); a01 = WMMA_F32(A0, b1, a01);
      a02 = WMMA_F32(A0, b2, a02); a03 = WMMA_F32(A0, b3, a03);
      a10 = WMMA_F32(A1, b0, a10); a11 = WMMA_F32(A1, b1, a11);
      a12 = WMMA_F32(A1, b2, a12); a13 = WMMA_F32(A1, b3, a13);
      woff += 4;
      bidx += 4 * N;
    }
  } else {            // -------- edge: clamped offsets --------
    int nj[4];
#pragma unroll
    for (int j = 0; j < 4; ++j) {
      int n = n0 + 16 * j + l16;
      nj[j] = (n > N - 1) ? (N - 1) : n;
    }
    int brow = (half * 2) * N;
    int woff = 0;
    for (int k0 = 0; k0 < K; k0 += 4) {
      v2f A0, A1;
      A0.x = Wm[woff];  A0.y = Wm[woff + 1];
      A1.x = Wm2[woff]; A1.y = Wm2[woff + 1];
      v2f b0, b1, b2, b3;
      b0.x = Zb[brow + nj[0]]; b0.y = Zb[brow + N + nj[0]];
      b1.x = Zb[brow + nj[1]]; b1.y = Zb[brow + N + nj[1]];
      b2.x = Zb[brow + nj[2]]; b2.y = Zb[brow + N + nj[2]];
      b3.x = Zb[brow + nj[3]]; b3.y = Zb[brow + N + nj[3]];
      a00 = WMMA_F32(A0, b0, a00); a01 = WMMA_F32(A0, b1, a01);
      a02 = WMMA_F32(A0, b2, a02); a03 = WMMA_F32(A0, b3, a03);
      a10 = WMMA_F32(A1, b0, a10); a11 = WMMA_F32(A1, b1, a11);
      a12 = WMMA_F32(A1, b2, a12); a13 = WMMA_F32(A1, b3, a13);
      woff += 4;
      brow += 4 * N;
    }
  }

  v8f* accs[8] = {&a00, &a01, &a02, &a03, &a10, &a11, &a12, &a13};
#pragma unroll
  for (int i = 0; i < 2; ++i) {
#pragma unroll
    for (int j = 0; j < 4; ++j) {
      int nn = n0 + 16 * j + l16;
      if (nn < N) {
#pragma unroll
        for (int r = 0; r < 8; ++r) {
          int mm = m0 + i * 16 + r + half * 8;
          Y[((long)b * M + mm) * N + nn] = (*accs[i * 4 + j])[r] + bias[mm];
        }
      }
    }
  }
}

// ---------------------------------------------------------------------------
// Fused GLU(channels) + residual + LayerNorm(channels) per (b,t) column.
// ---------------------------------------------------------------------------
__global__ void k_glu_res_ln(float* __restrict__ h, const float* __restrict__ y,
                             const float* __restrict__ w,
                             const float* __restrict__ bvec, int L) {
  __shared__ float s1[HDIM], s2[HDIM];
  int b = blockIdx.x / L, t = blockIdx.x % L;
  int c = threadIdx.x;
  long hidx = ((long)b * HDIM + c) * L + t;
  float u = y[((long)b * 2 * HDIM + c) * L + t];
  float g = y[((long)b * 2 * HDIM + HDIM + c) * L + t];
  float v = h[hidx] + u / (1.0f + __expf(-g));  // residual + GLU
  s1[c] = v;
  s2[c] = v * v;
  __syncthreads();
  for (int off = HDIM / 2; off > 0; off >>= 1) {
    if (c < off) {
      s1[c] += s1[c + off];
      s2[c] += s2[c + off];
    }
    __syncthreads();
  }
  float mu = s1[0] * (1.0f / HDIM);
  float var = s2[0] * (1.0f / HDIM) - mu * mu;
  float inv = rsqrtf(var + 1e-5f);
  h[hidx] = (v - mu) * inv * w[c] + bvec[c];
}

// ---------------------------------------------------------------------------
// Mean over channels -> x2 (B, L)
// ---------------------------------------------------------------------------
__global__ void k_chan_mean(const float* __restrict__ h, float* __restrict__ x2) {
  int idx = blockIdx.x * blockDim.x + threadIdx.x;
  if (idx >= BATCH * LSEQ) return;
  int b = idx / LSEQ, t = idx % LSEQ;
  float acc = 0.0f;
  for (int c = 0; c < HDIM; ++c) acc += h[((long)b * HDIM + c) * LSEQ + t];
  x2[idx] = acc * (1.0f / HDIM);
}

// ---------------------------------------------------------------------------
// Final linear (735x735) + per-output PReLU. Tiny (34 MFLOP) -> scalar.
// ---------------------------------------------------------------------------
__global__ void k_final_linear(const float* __restrict__ x2,
                               const float* __restrict__ W,
                               const float* __restrict__ bias,
                               const float* __restrict__ pa,
                               float* __restrict__ out) {
  int idx = blockIdx.x * blockDim.x + threadIdx.x;
  if (idx >= BATCH * LSEQ) return;
  int b = idx / LSEQ, o = idx % LSEQ;
  float acc = bias[o];
  const float* xr = x2 + b * LSEQ;
  const float* wr = W + (long)o * LSEQ;
  for (int t = 0; t < LSEQ; ++t) acc += xr[t] * wr[t];
  out[idx] = (acc >= 0.0f) ? acc : pa[o] * acc;
}

// ---------------------------------------------------------------------------
// Host-side launch
// ---------------------------------------------------------------------------
extern "C" void kernel_launch(void* const* d_in, const int* in_sizes, int n_in,
                              void* d_out, int out_size, void* d_ws, size_t ws_size,
                              hipStream_t stream) {
  const float* x        = (const float*)d_in[0];
  const float* low_hz   = (const float*)d_in[1];
  const float* band_hz  = (const float*)d_in[2];
  const float* w1       = (const float*)d_in[3];
  const float* w2       = (const float*)d_in[4];
  const float* w3       = (const float*)d_in[5];
  const float* w4       = (const float*)d_in[6];
  const float* w5       = (const float*)d_in[7];
  const float* w6       = (const float*)d_in[8];
  const float* prelu    = (const float*)d_in[9];
  const float* s4_logdt = (const float*)d_in[10];
  const float* s4_C     = (const float*)d_in[11];
  const float* s4_logA  = (const float*)d_in[12];
  const float* s4_Aimag = (const float*)d_in[13];
  const float* s4_D     = (const float*)d_in[14];
  const float* s4_out_w = (const float*)d_in[15];
  const float* s4_out_b = (const float*)d_in[16];
  const float* ln_w     = (const float*)d_in[17];
  const float* ln_b     = (const float*)d_in[18];
  const float* lin_w    = (const float*)d_in[19];
  const float* lin_b    = (const float*)d_in[20];
  const float* cprelu   = (const float*)d_in[21];
  const float* bnw[6];
  const float* bnb[6];
  for (int i = 0; i < 6; ++i) {
    bnw[i] = (const float*)d_in[22 + 2 * i];
    bnb[i] = (const float*)d_in[23 + 2 * i];
  }

  // workspace layout (floats)
  float* ws   = (float*)d_ws;
  float* filt = ws;                        // 4016 (pad to 4096)
  float* bufA = ws + 4096;                 // 15,360,000 (sinc out / conv / h)
  float* bufB = bufA + 15360000;           // 12,281,856 (conv ping / y)
  float* bufC = bufB + 12281856;           // 6,021,120  (z)
  float* Kbuf = bufC + 6021120;            // 188,160    (S4D kernel)
  float* x2   = Kbuf + 188160;             // 23,520
  size_t need = (size_t)(4096 + 15360000 + 12281856 + 6021120 + 188160 + 23520)
                * sizeof(float);
  if (ws_size < need) return;

  k_build_sinc_filters<<<(16 * KS + 255) / 256, 256, 0, stream>>>(low_hz, band_hz,
                                                                  filt);
  k_sinc_conv<<<dim3((30000 + 255) / 256, BATCH * 16), 256, 0, stream>>>(x, filt,
                                                                         bufA);

#define CONV_GRID(Cout, Lout) \
  ((BATCH * ((Cout) >> 5) * (((Lout) + 63) >> 6) + 7) / 8)

  k_conv_bn_prelu_wmma<20, 5><<<CONV_GRID(64, 5997), 256, 0, stream>>>(
      bufA, w1, bnw[0], bnb[0], prelu, 0, bufB, 16, 30000, 64, 5997);
  k_conv_bn_prelu_wmma<11, 2><<<CONV_GRID(64, 2994), 256, 0, stream>>>(
      bufB, w2, bnw[1], bnb[1], prelu, 1, bufA, 64, 5997, 64, 2994);
  k_conv_bn_prelu_wmma<11, 1><<<CONV_GRID(128, 2984), 256, 0, stream>>>(
      bufA, w3, bnw[2], bnb[2], prelu, 2, bufB, 64, 2994, 128, 2984);
  k_conv_bn_prelu_wmma<10, 2><<<CONV_GRID(128, 1488), 256, 0, stream>>>(
      bufB, w4, bnw[3], bnb[3], prelu, 3, bufA, 128, 2984, 128, 1488);
  k_conv_bn_prelu_wmma<11, 1><<<CONV_GRID(256, 1478), 256, 0, stream>>>(
      bufA, w5, bnw[4], bnb[4], prelu, 4, bufB, 128, 1488, 256, 1478);
  k_conv_bn_prelu_wmma<10, 2><<<CONV_GRID(256, 735), 256, 0, stream>>>(
      bufB, w6, bnw[5], bnb[5], prelu, 5, bufA, 256, 1478, 256, 735);

  for (int i = 0; i < NLAYERS; ++i) {
    k_s4d_build<<<HDIM, 256, 0, stream>>>(
        s4_logdt + i * HDIM, s4_C + (long)i * HDIM * NMODES * 2,
        s4_logA + (long)i * HDIM * NMODES, s4_Aimag + (long)i * HDIM * NMODES,
        Kbuf, LSEQ);
    k_causal_conv_gelu<<<BATCH * HDIM, 256, 0, stream>>>(bufA, Kbuf,
                                                         s4_D + i * HDIM, bufC,
                                                         LSEQ);
    int gtiles = BATCH * ((2 * HDIM) >> 5) * ((LSEQ + 63) >> 6);
    k_gemm_wmma_bias<<<(gtiles + 7) / 8, 256, 0, stream>>>(
        s4_out_w + (long)i * 2 * HDIM * HDIM, bufC, s4_out_b + i * 2 * HDIM,
        bufB, 2 * HDIM, HDIM, LSEQ, BATCH);
    k_glu_res_ln<<<BATCH * LSEQ, HDIM, 0, stream>>>(bufA, bufB, ln_w + i * HDIM,
                                                    ln_b + i * HDIM, LSEQ);
  }

  k_chan_mean<<<(BATCH * LSEQ + 255) / 256, 256, 0, stream>>>(bufA, x2);
  k_final_linear<<<(BATCH * LSEQ + 255) / 256, 256, 0, stream>>>(
      x2, lin_w, lin_b, cprelu, (float*)d_out);
}